// Model_70136815943749
// MI455X (gfx1250) — compile-verified
//
#include <hip/hip_runtime.h>

// GCN backbone for MI455X (gfx1250).
// Roofline: 13 GFLOP of GEMM (negligible) vs ~6.5 GB of edge gather/scatter
// traffic which fits in the 192MB L2 (h is 51MB) -> L2-atomic bound.
// Dense transforms use V_WMMA_F32_16X16X4_F32 (full fp32 precision; compute
// is nowhere near the bottleneck so no reason to drop to bf16/fp8).

#define NN 100000   // nodes
#define NE 1600000  // edges
#define DD 128      // feature dim (in == hid == out)

typedef __attribute__((ext_vector_type(2))) float v2f;
typedef __attribute__((ext_vector_type(8))) float v8f;

// ---- degree / normalization -------------------------------------------------
__global__ __launch_bounds__(256) void k_deg_init(float* __restrict__ deg) {
  int i = blockIdx.x * 256 + threadIdx.x;
  if (i < NN) deg[i] = 1.0f;  // self-loop contributes 1
}

__global__ __launch_bounds__(256) void k_deg_count(const int* __restrict__ dst,
                                                   float* __restrict__ deg) {
  int e = blockIdx.x * 256 + threadIdx.x;
  if (e < NE) atomicAdd(&deg[dst[e]], 1.0f);
}

__global__ __launch_bounds__(256) void k_rsqrt(float* __restrict__ d) {
  int i = blockIdx.x * 256 + threadIdx.x;
  if (i < NN) d[i] = rsqrtf(d[i]);
}

// ---- dense transform: H = X @ W, one wave per 16x16 output tile -------------
// f32 WMMA fragment layouts (ISA 7.12.2):
//   A 16x4:  lane<16 -> M=lane, {K0,K1}; lane>=16 -> M=lane-16, {K2,K3}
//   B 4x16:  v0: K=0 (lanes 0-15) / K=2 (lanes 16-31); v1: K=1 / K=3
//   C 16x16: vgpr r: M=r (lanes 0-15) / M=r+8 (lanes 16-31), N=lane%16
__global__ __launch_bounds__(256) void k_gemm_wmma(const float* __restrict__ X,
                                                   const float* __restrict__ W,
                                                   float* __restrict__ H) {
  const int wave = (blockIdx.x * 256 + (int)threadIdx.x) >> 5;
  const int lane = threadIdx.x & 31;
  const int row0 = (wave >> 3) << 4;  // DD/16 == 8 column tiles
  const int col0 = (wave & 7) << 4;
  if (row0 >= NN) return;             // wave-uniform guard
  const int half = lane >> 4;
  const int l16  = lane & 15;

  v8f acc = {};
  const float* xr = X + (size_t)(row0 + l16) * DD;  // my A row
  const float* wc = W + col0 + l16;                 // my B column
  for (int k = 0; k < DD; k += 4) {
    v2f a = *(const v2f*)(xr + k + 2 * half);       // 8B aligned (k%4==0)
    v2f b;
    b.x = wc[(size_t)(k + 2 * half) * DD];
    b.y = wc[(size_t)(k + 2 * half + 1) * DD];
    acc = __builtin_amdgcn_wmma_f32_16x16x4_f32(
        /*neg_a=*/false, a, /*neg_b=*/false, b,
        /*c_mod=*/(short)0, acc, /*reuse_a=*/false, /*reuse_b=*/false);
  }
  float* hp = H + (size_t)row0 * DD + col0 + l16;
#pragma unroll
  for (int r = 0; r < 8; ++r)
    hp[(size_t)(r + 8 * half) * DD] = acc[r];
}

// ---- zero scratch accumulator ----------------------------------------------
__global__ __launch_bounds__(256) void k_zero(float* __restrict__ p, int n) {
  int i = blockIdx.x * 256 + threadIdx.x;
  if (i < n) p[i] = 0.0f;
}

// ---- edge scatter: one wave per edge, 32 lanes x float4 = 512B row ----------
__global__ __launch_bounds__(256) void k_scatter(const int* __restrict__ src,
                                                 const int* __restrict__ dst,
                                                 const float* __restrict__ dinv,
                                                 const float* __restrict__ H,
                                                 float* __restrict__ agg) {
  const int e = (blockIdx.x * 256 + (int)threadIdx.x) >> 5;
  const int lane = threadIdx.x & 31;
  if (e >= NE) return;
  const int s = src[e];
  const int d = dst[e];
  const float norm = dinv[s] * dinv[d];
  const float4 v = ((const float4*)(H + (size_t)s * DD))[lane];  // coalesced 512B
  float* a = agg + (size_t)d * DD + lane * 4;
  atomicAdd(a + 0, v.x * norm);
  atomicAdd(a + 1, v.y * norm);
  atomicAdd(a + 2, v.z * norm);
  atomicAdd(a + 3, v.w * norm);
}

// ---- out = [relu](agg + h*dinv^2 + b) ---------------------------------------
__global__ __launch_bounds__(256) void k_finalize(const float* __restrict__ agg,
                                                  const float* __restrict__ H,
                                                  const float* __restrict__ dinv,
                                                  const float* __restrict__ bias,
                                                  float* __restrict__ out,
                                                  int relu) {
  int i = blockIdx.x * 256 + threadIdx.x;
  if (i >= NN * DD) return;
  int n = i >> 7;          // DD == 128
  int f = i & (DD - 1);
  float di = dinv[n];
  float v = agg[i] + H[i] * di * di + bias[f];
  out[i] = relu ? fmaxf(v, 0.0f) : v;
}

extern "C" void kernel_launch(void* const* d_in, const int* in_sizes, int n_in,
                              void* d_out, int out_size, void* d_ws, size_t ws_size,
                              hipStream_t stream) {
  // setup_inputs order: x1, edge_index1, x2, edge_index2, W0, b0, W1, b1
  const float* x[2]  = {(const float*)d_in[0], (const float*)d_in[2]};
  const int*   ei[2] = {(const int*)d_in[1],   (const int*)d_in[3]};
  const float* W0 = (const float*)d_in[4];
  const float* b0 = (const float*)d_in[5];
  const float* W1 = (const float*)d_in[6];
  const float* b1 = (const float*)d_in[7];
  float* out = (float*)d_out;

  // workspace layout (floats): dinv | H | AGG | Y  (reused across stages)
  float* ws   = (float*)d_ws;
  float* dinv = ws;
  float* H    = ws + 100352;                 // NN rounded up, keeps 16B align
  float* AG   = H  + (size_t)NN * DD;
  float* Y    = AG + (size_t)NN * DD;

  const int nd    = NN * DD;                       // 12.8M
  const dim3 B(256);
  const int gN    = (NN + 255) / 256;
  const int gE    = (NE + 255) / 256;
  const int gND   = (nd + 255) / 256;
  const int gGemm = (NN / 16) * (DD / 16) / 8;     // 8 waves per block
  const int gScat = NE / 8;                        // 8 edge-waves per block

  for (int bb = 0; bb < 2; ++bb) {
    const int* srcp = ei[bb];
    const int* dstp = ei[bb] + NE;

    // symmetric normalization deg^{-1/2}
    k_deg_init <<<gN, B, 0, stream>>>(dinv);
    k_deg_count<<<gE, B, 0, stream>>>(dstp, dinv);
    k_rsqrt    <<<gN, B, 0, stream>>>(dinv);

    // conv 1 (+ReLU)
    k_gemm_wmma<<<gGemm, B, 0, stream>>>(x[bb], W0, H);
    k_zero     <<<gND,   B, 0, stream>>>(AG, nd);
    k_scatter  <<<gScat, B, 0, stream>>>(srcp, dstp, dinv, H, AG);
    k_finalize <<<gND,   B, 0, stream>>>(AG, H, dinv, b0, Y, 1);

    // conv 2
    k_gemm_wmma<<<gGemm, B, 0, stream>>>(Y, W1, H);
    k_zero     <<<gND,   B, 0, stream>>>(AG, nd);
    k_scatter  <<<gScat, B, 0, stream>>>(srcp, dstp, dinv, H, AG);
    k_finalize <<<gND,   B, 0, stream>>>(AG, H, dinv, b1, out + (size_t)bb * nd, 0);
  }
}